// FastFFN_53841710022658
// MI455X (gfx1250) — compile-verified
//
#include <hip/hip_runtime.h>
#include <hip/hip_bf16.h>

typedef float v2f __attribute__((ext_vector_type(2)));
typedef float v8f __attribute__((ext_vector_type(8)));

#define DIMD   512
#define LEAFF  128
#define NLEAF  8
#define XPAD   516   // 512 + 4 -> LDS row stride, banks spaced by 4
#define HPAD   132   // 128 + 4

// ---------------------------------------------------------------------------
// Pass 0: zero the per-leaf counters (deterministic per launch / graph replay)
// ---------------------------------------------------------------------------
__global__ void zero_cnt_kernel(int* __restrict__ cnt) {
    if (threadIdx.x < NLEAF) cnt[threadIdx.x] = 0;
}

// ---------------------------------------------------------------------------
// Pass 1: routing. One wave32 per token. Hard decision = rint(sigmoid(s)),
// which is (s > 0) since sigmoid(0)=0.5 and rint(0.5)==0 (round-to-even).
// ---------------------------------------------------------------------------
__global__ __launch_bounds__(256)
void route_kernel(const float* __restrict__ x,
                  const float* __restrict__ wn0, const float* __restrict__ bn0,
                  const float* __restrict__ wn1, const float* __restrict__ bn1,
                  const float* __restrict__ wn2, const float* __restrict__ bn2,
                  int* __restrict__ cnt, int* __restrict__ bucket,
                  int n_tokens) {
    const int tok  = (blockIdx.x * blockDim.x + threadIdx.x) >> 5;
    const int lane = threadIdx.x & 31;
    if (tok >= n_tokens) return;

    const float* __restrict__ xr = x + (size_t)tok * DIMD;
    float a0 = 0.f, a1 = 0.f, a2 = 0.f, a3 = 0.f, a4 = 0.f, a5 = 0.f, a6 = 0.f;
    #pragma unroll 4
    for (int k = lane; k < DIMD; k += 32) {
        const float xv = xr[k];
        a0 = fmaf(xv, wn0[k], a0);
        const float2 w1 = *(const float2*)(wn1 + 2 * k);
        a1 = fmaf(xv, w1.x, a1);
        a2 = fmaf(xv, w1.y, a2);
        const float4 w2v = *(const float4*)(wn2 + 4 * k);
        a3 = fmaf(xv, w2v.x, a3);
        a4 = fmaf(xv, w2v.y, a4);
        a5 = fmaf(xv, w2v.z, a5);
        a6 = fmaf(xv, w2v.w, a6);
    }
    #pragma unroll
    for (int m = 16; m >= 1; m >>= 1) {
        a0 += __shfl_xor(a0, m, 32);
        a1 += __shfl_xor(a1, m, 32);
        a2 += __shfl_xor(a2, m, 32);
        a3 += __shfl_xor(a3, m, 32);
        a4 += __shfl_xor(a4, m, 32);
        a5 += __shfl_xor(a5, m, 32);
        a6 += __shfl_xor(a6, m, 32);
    }
    if (lane == 0) {
        const float s0 = a0 + bn0[0];
        const int   h0 = (s0 > 0.f) ? 1 : 0;
        const int   i1 = 1 - h0;                       // level-1 node
        const float s1 = (i1 == 0 ? a1 : a2) + bn1[i1];
        const int   h1 = (s1 > 0.f) ? 1 : 0;
        const int   i2 = i1 * 2 + (1 - h1);            // level-2 node
        float s2;
        switch (i2) {
            case 0:  s2 = a3 + bn2[0]; break;
            case 1:  s2 = a4 + bn2[1]; break;
            case 2:  s2 = a5 + bn2[2]; break;
            default: s2 = a6 + bn2[3]; break;
        }
        const int h2   = (s2 > 0.f) ? 1 : 0;
        const int leaf = i2 * 2 + (1 - h2);
        const int pos  = atomicAdd(&cnt[leaf], 1);
        bucket[leaf * n_tokens + pos] = tok;
    }
}

// ---------------------------------------------------------------------------
// Pass 2: per-leaf gated FFN on gathered 16-token tiles via f32 WMMA.
//   H = (X W1a + b1a) * (X W1b + b1b),  Y = H W2 + b2
// Block = 8 waves. Wave w owns F-tile [16w,16w+16) of H, then 4 D-tiles of Y.
// ---------------------------------------------------------------------------
__global__ __launch_bounds__(256)
void ffn_kernel(const float* __restrict__ x,
                const float* __restrict__ w1a, const float* __restrict__ b1a,
                const float* __restrict__ w1b, const float* __restrict__ b1b,
                const float* __restrict__ w2,  const float* __restrict__ b2,
                const int* __restrict__ cnt,   const int* __restrict__ bucket,
                float* __restrict__ out, int n_tokens) {
    __shared__ float xs[16 * XPAD];
    __shared__ float hs[16 * HPAD];
    __shared__ int   toks[16];
    __shared__ int   nvalid_s;

    const int leaf  = blockIdx.y;
    const int count = cnt[leaf];
    const int tile0 = blockIdx.x * 16;
    if (tile0 >= count) return;          // uniform per block: EXEC stays full

    const int tid = threadIdx.x;
    if (tid < 16) {
        const int idx = tile0 + tid;
        toks[tid] = (idx < count) ? bucket[leaf * n_tokens + idx]
                                  : bucket[leaf * n_tokens + tile0]; // pad row
    }
    if (tid == 0) nvalid_s = min(16, count - tile0);
    __syncthreads();
    const int nvalid = nvalid_s;

    // gather X tile (16 x 512) into LDS, float4, fully coalesced
    for (int i = tid; i < 16 * (DIMD / 4); i += 256) {
        const int row = i >> 7;
        const int c4  = (i & 127) << 2;
        const float4 v = *(const float4*)(x + (size_t)toks[row] * DIMD + c4);
        *(float4*)(xs + row * XPAD + c4) = v;
    }
    __syncthreads();

    const int wave  = tid >> 5;
    const int lane  = tid & 31;
    const int row16 = lane & 15;
    const int koff  = (lane >> 4) * 2;   // A/B frag: hi half-wave holds K+2..K+3
    const int rbase = (lane >> 4) * 8;   // C/D frag: hi half-wave holds M+8

    // hint: pull this wave's W2 panel toward the caches for phase 2
    __builtin_prefetch(w2 + (size_t)leaf * LEAFF * DIMD + wave * 64, 0, 1);

    // ---- phase 1: H tile ----
    {
        const int f0 = wave * 16;
        const float* __restrict__ Wa =
            w1a + (size_t)leaf * DIMD * LEAFF + f0 + row16;
        const float* __restrict__ Wb =
            w1b + (size_t)leaf * DIMD * LEAFF + f0 + row16;
        v8f ca = {};
        v8f cb = {};
        for (int k0 = 0; k0 < DIMD; k0 += 4) {
            v2f a;   // A: x[M=row16][k0+koff .. +1]  (8B-aligned LDS pair)
            a.x = xs[row16 * XPAD + k0 + koff];
            a.y = xs[row16 * XPAD + k0 + koff + 1];
            v2f ba, bb;  // B: w1[k][f], rows K, lane-contiguous in N=f
            ba.x = Wa[(size_t)(k0 + koff) * LEAFF];
            ba.y = Wa[(size_t)(k0 + koff + 1) * LEAFF];
            bb.x = Wb[(size_t)(k0 + koff) * LEAFF];
            bb.y = Wb[(size_t)(k0 + koff + 1) * LEAFF];
            ca = __builtin_amdgcn_wmma_f32_16x16x4_f32(false, a, false, ba,
                                                       (short)0, ca, false, false);
            cb = __builtin_amdgcn_wmma_f32_16x16x4_f32(false, a, false, bb,
                                                       (short)0, cb, false, false);
        }
        const float biasa = b1a[leaf * LEAFF + f0 + row16];
        const float biasb = b1b[leaf * LEAFF + f0 + row16];
        #pragma unroll
        for (int i = 0; i < 8; ++i) {
            const float h = (ca[i] + biasa) * (cb[i] + biasb);  // gate
            hs[(i + rbase) * HPAD + f0 + row16] = h;
        }
    }
    __syncthreads();

    // ---- phase 2: Y tiles ----
    #pragma unroll
    for (int j = 0; j < 4; ++j) {
        const int d0 = wave * 64 + j * 16;
        const float* __restrict__ W =
            w2 + (size_t)leaf * LEAFF * DIMD + d0 + row16;
        v8f acc = {};
        for (int k0 = 0; k0 < LEAFF; k0 += 4) {
            v2f a;
            a.x = hs[row16 * HPAD + k0 + koff];
            a.y = hs[row16 * HPAD + k0 + koff + 1];
            v2f b;
            b.x = W[(size_t)(k0 + koff) * DIMD];
            b.y = W[(size_t)(k0 + koff + 1) * DIMD];
            acc = __builtin_amdgcn_wmma_f32_16x16x4_f32(false, a, false, b,
                                                        (short)0, acc, false, false);
        }
        const float bias = b2[leaf * DIMD + d0 + row16];
        #pragma unroll
        for (int i = 0; i < 8; ++i) {
            const int r = i + rbase;
            if (r < nvalid) {
                out[(size_t)toks[r] * DIMD + d0 + row16] = acc[i] + bias;
            }
        }
    }
}

// ---------------------------------------------------------------------------
extern "C" void kernel_launch(void* const* d_in, const int* in_sizes, int n_in,
                              void* d_out, int out_size, void* d_ws, size_t ws_size,
                              hipStream_t stream) {
    const float* inputs = (const float*)d_in[0];
    // d_in[1] = training (forward value identical either way; ignored)
    const float* wn0 = (const float*)d_in[2];
    const float* bn0 = (const float*)d_in[3];
    const float* wn1 = (const float*)d_in[4];
    const float* bn1 = (const float*)d_in[5];
    const float* wn2 = (const float*)d_in[6];
    const float* bn2 = (const float*)d_in[7];
    const float* w1a = (const float*)d_in[8];
    const float* b1a = (const float*)d_in[9];
    const float* w1b = (const float*)d_in[10];
    const float* b1b = (const float*)d_in[11];
    const float* w2  = (const float*)d_in[12];
    const float* b2  = (const float*)d_in[13];
    float* out = (float*)d_out;

    const int n_tokens = in_sizes[0] / DIMD;   // B*T = 8192

    int* cnt    = (int*)d_ws;                  // 8 counters
    int* bucket = cnt + 16;                    // 8 * n_tokens token ids

    zero_cnt_kernel<<<1, 32, 0, stream>>>(cnt);

    const int route_blocks = (n_tokens * 32 + 255) / 256;
    route_kernel<<<route_blocks, 256, 0, stream>>>(
        inputs, wn0, bn0, wn1, bn1, wn2, bn2, cnt, bucket, n_tokens);

    dim3 grid((n_tokens + 15) / 16, NLEAF);
    ffn_kernel<<<grid, 256, 0, stream>>>(
        inputs, w1a, b1a, w1b, b1b, w2, b2, cnt, bucket, out, n_tokens);
}